// LightGCNModel_64458869179103
// MI455X (gfx1250) — compile-verified
//
#include <hip/hip_runtime.h>
#include <hip/hip_bf16.h>

#define NUM_USERS 100000
#define NUM_ITEMS 50000
#define DIM 64
#define N_NODES (NUM_USERS + NUM_ITEMS)
#define E0 600000
#define BATCH 8192
#define L_W 1e-4f

typedef __attribute__((ext_vector_type(2))) float v2f;
typedef __attribute__((ext_vector_type(8))) float v8f;

// ---------------- utility kernels ----------------

__global__ __launch_bounds__(256) void zero_f32(float* p, long long n) {
    long long i = (long long)blockIdx.x * 256 + threadIdx.x;
    if (i < n) p[i] = 0.0f;
}

__global__ __launch_bounds__(256) void init_h(const float* __restrict__ Gu,
                                              const float* __restrict__ Gi,
                                              float* __restrict__ h,
                                              float* __restrict__ acc) {
    long long i = (long long)blockIdx.x * 256 + threadIdx.x;
    const long long NU = (long long)NUM_USERS * DIM;
    const long long NT = (long long)N_NODES * DIM;
    if (i >= NT) return;
    float v = (i < NU) ? Gu[i] : Gi[i - NU];
    h[i] = v;
    acc[i] = v;
}

__global__ __launch_bounds__(256) void deg_kernel(const int* __restrict__ eu,
                                                  const int* __restrict__ ei,
                                                  float* __restrict__ deg) {
    int e = blockIdx.x * 256 + threadIdx.x;
    if (e >= E0) return;
    atomicAdd(&deg[eu[e]], 1.0f);
    atomicAdd(&deg[NUM_USERS + ei[e]], 1.0f);
}

__global__ __launch_bounds__(256) void dinv_kernel(float* __restrict__ deg) {
    int n = blockIdx.x * 256 + threadIdx.x;
    if (n >= N_NODES) return;
    float d = deg[n];
    deg[n] = (d > 0.0f) ? rsqrtf(d) : 0.0f;   // in-place: deg becomes dinv
}

__global__ __launch_bounds__(256) void w_kernel(const int* __restrict__ eu,
                                                const int* __restrict__ ei,
                                                const float* __restrict__ dinv,
                                                float* __restrict__ w) {
    int e = blockIdx.x * 256 + threadIdx.x;
    if (e >= E0) return;
    w[e] = dinv[eu[e]] * dinv[NUM_USERS + ei[e]];
}

// 64 threads per raw edge: thread handles one dim d for both directions.
// Loads of h[row*64 + d] are coalesced 256B rows; scatter via f32 atomics (L2 RMW).
__global__ __launch_bounds__(256) void scatter_kernel(const int* __restrict__ eu,
                                                      const int* __restrict__ ei,
                                                      const float* __restrict__ w,
                                                      const float* __restrict__ h,
                                                      float* __restrict__ hn) {
    long long tid = (long long)blockIdx.x * 256 + threadIdx.x;
    int e = (int)(tid >> 6);
    int d = (int)(tid & 63);
    if (e >= E0) return;
    int u  = eu[e];
    int it = ei[e] + NUM_USERS;
    float we = w[e];
    float hu = h[(size_t)u  * DIM + d];
    float hi = h[(size_t)it * DIM + d];
    atomicAdd(&hn[(size_t)it * DIM + d], we * hu);
    atomicAdd(&hn[(size_t)u  * DIM + d], we * hi);
}

__global__ __launch_bounds__(256) void acc_add(float* __restrict__ acc,
                                               const float* __restrict__ h) {
    long long i = (long long)blockIdx.x * 256 + threadIdx.x;
    if (i < (long long)N_NODES * DIM) acc[i] += h[i];
}

__global__ void zero_out(float* out) { out[0] = 0.0f; }

// ---------------- BPR loss with fp32 WMMA ----------------
// One wave per 16 batch rows. Accumulate 16x16 cross-dot matrices for (u,pos)
// and (u,neg) with V_WMMA_F32_16X16X4_F32 over K=64 in 16 steps; diagonal gives
// the per-row dot products. Sum-of-squares for the L2 reg is folded into the
// same loads (each element of the 16-row blocks is loaded exactly once per wave).
__global__ __launch_bounds__(256) void bpr_wmma_kernel(const float* __restrict__ acc,
                                                       const int* __restrict__ user,
                                                       const int* __restrict__ pos,
                                                       const int* __restrict__ neg,
                                                       float* __restrict__ out) {
    int lane = threadIdx.x & 31;
    int wave = threadIdx.x >> 5;
    int group = blockIdx.x * 8 + wave;          // 512 groups of 16 rows
    int base = group * 16;
    int row  = lane & 15;
    int half = lane >> 4;                       // 0: K=0,1   1: K=2,3

    int u = user[base + row];
    int p = pos[base + row] + NUM_USERS;
    int n = neg[base + row] + NUM_USERS;
    const float* gu = acc + (size_t)u * DIM;
    const float* gp = acc + (size_t)p * DIM;
    const float* gn = acc + (size_t)n * DIM;

    const float scale = 0.25f;                  // emb = acc / (N_LAYERS + 1)
    v8f cpos = {};
    v8f cneg = {};
    float ssq = 0.0f;

#pragma unroll
    for (int kc = 0; kc < DIM; kc += 4) {
        int koff = kc + 2 * half;
        v2f a, bp, bn;
        a.x  = gu[koff]     * scale;
        a.y  = gu[koff + 1] * scale;
        bp.x = gp[koff]     * scale;
        bp.y = gp[koff + 1] * scale;
        bn.x = gn[koff]     * scale;
        bn.y = gn[koff + 1] * scale;

        ssq += a.x * a.x + a.y * a.y
             + bp.x * bp.x + bp.y * bp.y
             + bn.x * bn.x + bn.y * bn.y;

        // D[m][n] += sum_k A[m][k] * B[k][n]   (full fp32, RNE)
        cpos = __builtin_amdgcn_wmma_f32_16x16x4_f32(false, a, false, bp,
                                                     (short)0, cpos, false, false);
        cneg = __builtin_amdgcn_wmma_f32_16x16x4_f32(false, a, false, bn,
                                                     (short)0, cneg, false, false);
    }

    // Diagonal extraction: (m,m) at lane m / VGPR m for m<8; lane m+16 / VGPR m-8 for m>=8.
    bool have = (lane < 8) || (lane >= 24);
    int idx = (lane < 8) ? lane : (lane - 24);
    float dp = 0.0f, dn = 0.0f;
    if (have) {
#pragma unroll
        for (int i = 0; i < 8; ++i) {
            if (idx == i) { dp = cpos[i]; dn = cneg[i]; }
        }
    }

    float term = 0.0f;
    if (have) {
        // -log_sigmoid(dp - dn) = softplus(dn - dp), numerically stable
        float x = dn - dp;
        term = fmaxf(x, 0.0f) + log1pf(expf(-fabsf(x)));
    }

    float partial = term * (1.0f / BATCH) + ssq * (L_W * 0.5f / BATCH);

    // wave32 reduction
#pragma unroll
    for (int off = 16; off > 0; off >>= 1)
        partial += __shfl_xor(partial, off, 32);

    if (lane == 0) atomicAdd(out, partial);
}

// ---------------- host launcher ----------------

extern "C" void kernel_launch(void* const* d_in, const int* in_sizes, int n_in,
                              void* d_out, int out_size, void* d_ws, size_t ws_size,
                              hipStream_t stream) {
    const float* Gu  = (const float*)d_in[0];
    const float* Gi  = (const float*)d_in[1];
    const int* eu    = (const int*)d_in[2];
    const int* ei    = (const int*)d_in[3];
    const int* user  = (const int*)d_in[4];
    const int* pos   = (const int*)d_in[5];
    const int* neg   = (const int*)d_in[6];
    float* out       = (float*)d_out;

    const long long ND = (long long)N_NODES * DIM;   // 9.6M floats

    // workspace layout (floats): deg/dinv | w | h0 | h1 | acc   (~118 MB)
    float* deg  = (float*)d_ws;
    float* w    = deg + N_NODES;
    float* h0   = w + E0;
    float* h1   = h0 + ND;
    float* accb = h1 + ND;

    const int T = 256;
    int b_nodes = (N_NODES + T - 1) / T;
    int b_edges = (E0 + T - 1) / T;
    int b_nd    = (int)((ND + T - 1) / T);
    int b_scat  = (int)(((long long)E0 * 64 + T - 1) / T);

    // 1) degrees
    zero_f32<<<b_nodes, T, 0, stream>>>(deg, N_NODES);
    deg_kernel<<<b_edges, T, 0, stream>>>(eu, ei, deg);
    dinv_kernel<<<b_nodes, T, 0, stream>>>(deg);
    w_kernel<<<b_edges, T, 0, stream>>>(eu, ei, deg, w);

    // 2) init h0 = acc = [Gu; Gi]
    init_h<<<b_nd, T, 0, stream>>>(Gu, Gi, h0, accb);

    // 3) three propagation layers (ping-pong)
    float* hcur = h0;
    float* hnext = h1;
    for (int l = 0; l < 3; ++l) {
        zero_f32<<<b_nd, T, 0, stream>>>(hnext, ND);
        scatter_kernel<<<b_scat, T, 0, stream>>>(eu, ei, w, hcur, hnext);
        acc_add<<<b_nd, T, 0, stream>>>(accb, hnext);
        float* t = hcur; hcur = hnext; hnext = t;
    }

    // 4) BPR loss (WMMA fp32) -> single scalar
    zero_out<<<1, 1, 0, stream>>>(out);
    bpr_wmma_kernel<<<BATCH / (16 * 8), T, 0, stream>>>(accb, user, pos, neg, out);
}